// RGATLayer_70617852281333
// MI455X (gfx1250) — compile-verified
//
#include <hip/hip_runtime.h>
#include <hip/hip_bf16.h>
#include <math.h>

typedef __attribute__((ext_vector_type(2))) float v2f;
typedef __attribute__((ext_vector_type(8))) float v8f;

#define FIN   256
#define FOUT  128
#define NHEAD 4
#define DHEAD 32
#define NEG_SLOPE 0.2f

// ---------------------------------------------------------------------------
// GEMM: h[N,128] = x[N,256] @ W[256,128] using V_WMMA_F32_16X16X4_F32.
// Block = 256 threads = 8 waves. Block owns 16 rows; wave w owns cols [16w,16w+16).
// A frag (16x4): lane l -> M = l&15, K = k0 + 2*(l>>4) + {0,1}  (2 VGPRs)
// B frag (4x16): lane l -> N = l&15, K = k0 + 2*(l>>4) + {0,1}  (2 VGPRs)
// C/D (16x16):   VGPR v, lane l -> M = v + 8*(l>>4), N = l&15   (8 VGPRs)
// ---------------------------------------------------------------------------
__global__ void __launch_bounds__(256)
rgat_gemm_wmma(const float* __restrict__ x, const float* __restrict__ Wr,
               float* __restrict__ h, int N) {
  const int lane  = threadIdx.x & 31;
  const int wave  = threadIdx.x >> 5;        // 0..7 : N-tile within 128 cols
  const int tm    = blockIdx.x;              // M-tile (16 rows)
  const int mrow  = lane & 15;
  const int kb    = (lane >> 4) << 1;        // 0 or 2
  const int col   = wave * 16 + mrow;        // B/N column for this lane

  int row = tm * 16 + mrow;
  int rowc = row < N ? row : 0;              // clamp loads; store handled below
  const float* __restrict__ xrow = x + (size_t)rowc * FIN;

  v8f c = {};
  for (int k0 = 0; k0 < FIN; k0 += 4) {
    v2f a = *(const v2f*)(xrow + k0 + kb);                 // 8B-aligned (kb even)
    v2f b;
    b.x = Wr[(size_t)(k0 + kb)     * FOUT + col];
    b.y = Wr[(size_t)(k0 + kb + 1) * FOUT + col];
    c = __builtin_amdgcn_wmma_f32_16x16x4_f32(
        /*neg_a=*/false, a, /*neg_b=*/false, b,
        /*c_mod=*/(short)0, c, /*reuse_a=*/false, /*reuse_b=*/false);
  }

  const int rbase = tm * 16 + 8 * (lane >> 4);
  const int ocol  = wave * 16 + (lane & 15);
  float* __restrict__ hp = h + (size_t)rbase * FOUT + ocol;

  if (tm * 16 + 16 <= N) {
    // Block-uniform fast path: tile fully in range, straight-line stores.
#pragma unroll
    for (int v = 0; v < 8; ++v) hp[(size_t)v * FOUT] = c[v];
  } else {
#pragma unroll
    for (int v = 0; v < 8; ++v) {
      if (rbase + v < N) hp[(size_t)v * FOUT] = c[v];
    }
  }
}

// ---------------------------------------------------------------------------
// Per-node attention halves: als/ald[N,4] = <h[n, head*32:..], a_src/a_dst[head]>
// ---------------------------------------------------------------------------
__global__ void rgat_node_logits(const float* __restrict__ h,
                                 const float* __restrict__ asrc,
                                 const float* __restrict__ adst,
                                 float* __restrict__ als, float* __restrict__ ald,
                                 int N) {
  int idx = blockIdx.x * blockDim.x + threadIdx.x;
  if (idx >= N * NHEAD) return;
  int n  = idx >> 2;
  int hd = idx & 3;
  const float* hp = h + (size_t)n * FOUT + hd * DHEAD;
  const float* as = asrc + hd * DHEAD;
  const float* ad = adst + hd * DHEAD;
  float s = 0.f, d = 0.f;
#pragma unroll
  for (int i = 0; i < DHEAD; ++i) {
    float v = hp[i];
    s = fmaf(v, as[i], s);
    d = fmaf(v, ad[i], d);
  }
  als[idx] = s;
  ald[idx] = d;
}

__global__ void rgat_init_out(const float* __restrict__ bias, float* __restrict__ out,
                              int total) {
  int i = blockIdx.x * blockDim.x + threadIdx.x;
  if (i >= total) return;
  int f = i & (FOUT - 1);
  out[i] = bias[f] + bias[FOUT + f] + bias[2 * FOUT + f];
}

__global__ void rgat_init_seg(float* __restrict__ amax, float* __restrict__ denom,
                              int nh) {
  int i = blockIdx.x * blockDim.x + threadIdx.x;
  if (i >= nh) return;
  amax[i]  = -1e30f;
  denom[i] = 0.f;
}

__device__ __forceinline__ float atomicMaxF(float* addr, float val) {
  // IEEE-ordered atomic max via signed/unsigned integer monotonicity.
  if (val >= 0.f)
    return __int_as_float(atomicMax((int*)addr, __float_as_int(val)));
  return __uint_as_float(atomicMin((unsigned int*)addr, __float_as_uint(val)));
}

__device__ __forceinline__ float leaky(float v) {
  return v > 0.f ? v : NEG_SLOPE * v;
}

// Pass 1: alpha = leakyrelu(als[src] + ald[dst]); segment-max into amax[dst].
__global__ void rgat_edge_alpha_max(const int* __restrict__ ei,
                                    const float* __restrict__ als,
                                    const float* __restrict__ ald,
                                    float* __restrict__ alpha,
                                    float* __restrict__ amax, int E) {
  int e = blockIdx.x * blockDim.x + threadIdx.x;
  if (e >= E) return;
  int s = ei[e];
  int d = ei[E + e];
  float4 a_s = *(const float4*)(als + (size_t)s * NHEAD);
  float4 a_d = *(const float4*)(ald + (size_t)d * NHEAD);
  float4 al;
  al.x = leaky(a_s.x + a_d.x);
  al.y = leaky(a_s.y + a_d.y);
  al.z = leaky(a_s.z + a_d.z);
  al.w = leaky(a_s.w + a_d.w);
  *(float4*)(alpha + (size_t)e * NHEAD) = al;
  float* am = amax + (size_t)d * NHEAD;
  atomicMaxF(am + 0, al.x);
  atomicMaxF(am + 1, al.y);
  atomicMaxF(am + 2, al.z);
  atomicMaxF(am + 3, al.w);
}

// Pass 2: e = exp(alpha - amax[dst]); segment-sum into denom[dst]; alpha := e.
__global__ void rgat_edge_exp_sum(const int* __restrict__ ei,
                                  float* __restrict__ alpha,
                                  const float* __restrict__ amax,
                                  float* __restrict__ denom, int E) {
  int e = blockIdx.x * blockDim.x + threadIdx.x;
  if (e >= E) return;
  int d = ei[E + e];
  float4 al = *(const float4*)(alpha + (size_t)e * NHEAD);
  float4 mx = *(const float4*)(amax + (size_t)d * NHEAD);
  al.x = __expf(al.x - mx.x);
  al.y = __expf(al.y - mx.y);
  al.z = __expf(al.z - mx.z);
  al.w = __expf(al.w - mx.w);
  *(float4*)(alpha + (size_t)e * NHEAD) = al;
  float* dn = denom + (size_t)d * NHEAD;
  atomicAdd(dn + 0, al.x);
  atomicAdd(dn + 1, al.y);
  atomicAdd(dn + 2, al.z);
  atomicAdd(dn + 3, al.w);
}

// Pass 3: one wave per edge; lane owns features {head*32 + lane} so the head
// weight is uniform per chunk. out[dst] += h[src] * w. h/out are L2-resident.
__global__ void __launch_bounds__(256)
rgat_edge_scatter(const int* __restrict__ ei, const float* __restrict__ h,
                  const float* __restrict__ alpha, const float* __restrict__ denom,
                  float* __restrict__ out, int E) {
  const int lane = threadIdx.x & 31;
  const int wave = threadIdx.x >> 5;
  int e = blockIdx.x * 8 + wave;
  if (e >= E) return;
  int s = ei[e];
  int d = ei[E + e];

  // Prefetch the source row of edge e+8, which the next block's wave gathers (gfx1250 path).
  if (e + 8 < E) {
    int s2 = ei[e + 8];
    __builtin_prefetch(h + (size_t)s2 * FOUT + lane, 0, 0);
  }

  float4 ev = *(const float4*)(alpha + (size_t)e * NHEAD);
  float4 dn = *(const float4*)(denom + (size_t)d * NHEAD);
  float w0 = ev.x / (dn.x + 1e-16f);
  float w1 = ev.y / (dn.y + 1e-16f);
  float w2 = ev.z / (dn.z + 1e-16f);
  float w3 = ev.w / (dn.w + 1e-16f);

  const float* __restrict__ hs = h + (size_t)s * FOUT;
  float* __restrict__ od = out + (size_t)d * FOUT;
  atomicAdd(od + 0 * DHEAD + lane, hs[0 * DHEAD + lane] * w0);
  atomicAdd(od + 1 * DHEAD + lane, hs[1 * DHEAD + lane] * w1);
  atomicAdd(od + 2 * DHEAD + lane, hs[2 * DHEAD + lane] * w2);
  atomicAdd(od + 3 * DHEAD + lane, hs[3 * DHEAD + lane] * w3);
}

// ---------------------------------------------------------------------------
extern "C" void kernel_launch(void* const* d_in, const int* in_sizes, int n_in,
                              void* d_out, int out_size, void* d_ws, size_t ws_size,
                              hipStream_t stream) {
  const float* x        = (const float*)d_in[0];
  const int*   edges[3] = {(const int*)d_in[1], (const int*)d_in[2], (const int*)d_in[3]};
  const float* W        = (const float*)d_in[4];
  const float* att_src  = (const float*)d_in[5];
  const float* att_dst  = (const float*)d_in[6];
  const float* bias     = (const float*)d_in[7];

  const int N = in_sizes[0] / FIN;      // 50000
  const int E = in_sizes[1] / 2;        // 800000
  float* out = (float*)d_out;

  // Workspace carve-up (reused across relations), 256B aligned slices.
  char* ws = (char*)d_ws;
  size_t off = 0;
  auto carve = [&](size_t bytes) {
    void* p = ws + off;
    off += (bytes + 255) & ~(size_t)255;
    return p;
  };
  float* h     = (float*)carve((size_t)N * FOUT * sizeof(float));
  float* als   = (float*)carve((size_t)N * NHEAD * sizeof(float));
  float* ald   = (float*)carve((size_t)N * NHEAD * sizeof(float));
  float* alpha = (float*)carve((size_t)E * NHEAD * sizeof(float));
  float* amax  = (float*)carve((size_t)N * NHEAD * sizeof(float));
  float* denom = (float*)carve((size_t)N * NHEAD * sizeof(float));
  (void)ws_size;

  // out = sum_r bias[r] (relations then scatter-accumulate into it)
  rgat_init_out<<<(N * FOUT + 255) / 256, 256, 0, stream>>>(bias, out, N * FOUT);

  for (int r = 0; r < 3; ++r) {
    const float* Wr = W + (size_t)r * FIN * FOUT;
    rgat_gemm_wmma<<<(N + 15) / 16, 256, 0, stream>>>(x, Wr, h, N);
    rgat_node_logits<<<(N * NHEAD + 255) / 256, 256, 0, stream>>>(
        h, att_src + (size_t)r * NHEAD * DHEAD, att_dst + (size_t)r * NHEAD * DHEAD,
        als, ald, N);
    rgat_init_seg<<<(N * NHEAD + 255) / 256, 256, 0, stream>>>(amax, denom, N * NHEAD);
    rgat_edge_alpha_max<<<(E + 255) / 256, 256, 0, stream>>>(
        edges[r], als, ald, alpha, amax, E);
    rgat_edge_exp_sum<<<(E + 255) / 256, 256, 0, stream>>>(
        edges[r], alpha, amax, denom, E);
    rgat_edge_scatter<<<(E + 7) / 8, 256, 0, stream>>>(
        edges[r], h, alpha, denom, out, E);
  }
}